// NEPAM_24283745091988
// MI455X (gfx1250) — compile-verified
//
#include <hip/hip_runtime.h>
#include <stdint.h>

// Problem constants (match reference)
#define B_    128
#define C_    768
#define HW_   576          // 24*24
#define TOK_  144          // 12*12 blocks
#define NSEL_ 360          // 72 merged pixels + 72*4 kept pixels
#define CCH_  32           // channels per gather LDS tile
#define NCC_  (C_ / CCH_)  // 24 tiles
#define LSTR_ 578          // padded floats per channel row in LDS (bank-conflict-free, 8B aligned)

typedef int v2i32 __attribute__((vector_size(8)));
typedef __attribute__((address_space(1))) v2i32 g_v2i32;   // global (device) AS
typedef __attribute__((address_space(3))) v2i32 l_v2i32;   // LDS AS

__device__ __forceinline__ void async_load_b64(const void* gptr, void* lptr) {
#if __has_builtin(__builtin_amdgcn_global_load_async_to_lds_b64)
  __builtin_amdgcn_global_load_async_to_lds_b64((g_v2i32*)gptr, (l_v2i32*)lptr, 0, 0);
#else
  asm volatile("global_load_async_to_lds_b64 %0, %1, off"
               :
               : "v"((uint32_t)(uintptr_t)(l_v2i32*)lptr),
                 "v"((uint64_t)(uintptr_t)gptr)
               : "memory");
#endif
}

__device__ __forceinline__ void wait_async0() {
#if __has_builtin(__builtin_amdgcn_s_wait_asynccnt)
  __builtin_amdgcn_s_wait_asynccnt(0);
#else
  asm volatile("s_wait_asynccnt 0x0" ::: "memory");
#endif
}

// ---------------------------------------------------------------------------
// K0: zero the score scratch (d_ws is poisoned; must init every launch)
// ---------------------------------------------------------------------------
__global__ __launch_bounds__(256) void nepam_zero(float* p, int n) {
  int i = blockIdx.x * 256 + threadIdx.x;
  if (i < n) p[i] = 0.0f;
}

// ---------------------------------------------------------------------------
// K1: score[b,t] = sum_c sum_{g=1..3} |x[b,c,pix(g,t)] - x[b,c,pix(0,t)]|
// grid = B_*8 blocks (96 channels each), 256 threads.
// Consecutive threads sweep tokens -> coalesced float2 row reads.
// ---------------------------------------------------------------------------
__global__ __launch_bounds__(256) void nepam_score(const float* __restrict__ x,
                                                   float* __restrict__ score_ws) {
  __shared__ float acc[TOK_];
  const int b     = blockIdx.x >> 3;
  const int chunk = blockIdx.x & 7;           // 8 chunks of 96 channels
  const int tid   = threadIdx.x;
  if (tid < TOK_) acc[tid] = 0.0f;
  __syncthreads();

  const float* xb = x + ((size_t)b * C_ + (size_t)chunk * 96) * HW_;
  const int PAIRS = 96 * TOK_;                // (channel-local, token) pairs
  for (int idx = tid; idx < PAIRS; idx += 256) {
    const int cl = idx / TOK_;
    const int t  = idx - cl * TOK_;
    const int a  = t / 12;                    // block row
    const int cb = t - a * 12;                // block col
    const float* plane = xb + (size_t)cl * HW_;
    const float2 v0 = *(const float2*)(plane + 48 * a + 2 * cb);        // row 2a
    const float2 v1 = *(const float2*)(plane + 48 * a + 24 + 2 * cb);   // row 2a+1
    const float r  = v0.x;                    // top-left (g=0) reference
    const float sv = fabsf(v0.y - r) + fabsf(v1.x - r) + fabsf(v1.y - r);
    atomicAdd(&acc[t], sv);                   // ds_add_f32
  }
  __syncthreads();
  if (tid < TOK_) atomicAdd(&score_ws[b * TOK_ + tid], acc[tid]);
}

// ---------------------------------------------------------------------------
// K2: per-batch stable argsort (rank-by-counting), token selection, and
// counting sort of the 360 (distinct) selected pixel indices.
// grid = B_, 512 threads.
// ---------------------------------------------------------------------------
__global__ __launch_bounds__(512) void nepam_select(const float* __restrict__ score_ws,
                                                    int* __restrict__ tok_ws,
                                                    float* __restrict__ tok_out) {
  __shared__ float s[TOK_];
  __shared__ int order[TOK_];
  __shared__ int cand[NSEL_];
  __shared__ int sorted_[NSEL_];
  const int b = blockIdx.x, tid = threadIdx.x;

  if (tid < TOK_) s[tid] = score_ws[b * TOK_ + tid];
  __syncthreads();

  // stable ascending argsort: rank = #{j: s[j] < s[i] || (s[j]==s[i] && j<i)}
  if (tid < TOK_) {
    const float si = s[tid];
    int cnt = 0;
    for (int j = 0; j < TOK_; ++j) {
      const float sj = s[j];
      cnt += (sj < si) || (sj == si && j < tid);
    }
    order[cnt] = tid;
  }
  __syncthreads();

  // build candidate pixel list: order[0:72] -> top-left only; order[72:] -> all 4
  if (tid < TOK_) {
    const int tokv = order[tid];
    const int a = tokv / 12, cb = tokv - 12 * (tokv / 12);
    const int p00 = 48 * a + 2 * cb;          // (2a)*24 + 2cb
    if (tid < 72) {
      cand[tid] = p00;
    } else {
      const int base = 72 + (tid - 72) * 4;
      cand[base + 0] = p00;
      cand[base + 1] = p00 + 1;
      cand[base + 2] = p00 + 24;
      cand[base + 3] = p00 + 25;
    }
  }
  __syncthreads();

  // counting sort (all 360 values distinct)
  if (tid < NSEL_) {
    const int v = cand[tid];
    int cnt = 0;
    for (int j = 0; j < NSEL_; ++j) cnt += (cand[j] < v);
    sorted_[cnt] = v;
  }
  __syncthreads();

  if (tid < NSEL_) {
    const int v = sorted_[tid];
    tok_ws[b * NSEL_ + tid]  = v;
    tok_out[b * NSEL_ + tid] = (float)v;      // second reference output
  }
}

// ---------------------------------------------------------------------------
// K3: out[b,t,c] = x[b,c,tok[b,t]]  via async global->LDS staging + LDS
// transpose. grid = (B_, NCC_), 256 threads, dynamic LDS = CCH_*LSTR_*4 B.
// Column reads hit banks (2*lane + p) % 64 -> conflict-free (stride 578).
// ---------------------------------------------------------------------------
__global__ __launch_bounds__(256) void nepam_gather(const float* __restrict__ x,
                                                    const int* __restrict__ tok,
                                                    float* __restrict__ out) {
  extern __shared__ float tile[];             // [CCH_][LSTR_]
  const int b   = blockIdx.x;
  const int c0  = blockIdx.y * CCH_;
  const int tid = threadIdx.x;

  const char* src = (const char*)(x + ((size_t)b * C_ + c0) * HW_);
  // async copy CCH_*576 floats, 8 bytes per lane-op
  const int NQ8 = CCH_ * HW_ / 2;             // 9216 b64 transfers
  for (int q8 = tid; q8 < NQ8; q8 += 256) {
    const int cl = q8 / (HW_ / 2);            // channel-local row
    const int r8 = q8 - cl * (HW_ / 2);       // b64 index within row
    async_load_b64(src + (size_t)q8 * 8,
                   (char*)tile + (size_t)cl * (LSTR_ * 4) + (size_t)r8 * 8);
  }
  wait_async0();
  __syncthreads();

  // wave-per-token writes: 32 lanes = 32 channels -> 128B coalesced stores
  const int lane = tid & 31;
  const int wv   = tid >> 5;                  // 8 waves
  const int* tb  = tok + b * NSEL_;
  float* ob      = out + (size_t)b * NSEL_ * C_ + c0;
  for (int t = wv; t < NSEL_; t += 8) {
    const int p = tb[t];
    ob[(size_t)t * C_ + lane] = tile[lane * LSTR_ + p];
  }
}

// ---------------------------------------------------------------------------
extern "C" void kernel_launch(void* const* d_in, const int* in_sizes, int n_in,
                              void* d_out, int out_size, void* d_ws, size_t ws_size,
                              hipStream_t stream) {
  const float* x = (const float*)d_in[0];
  float* out     = (float*)d_out;
  float* tok_out = out + (size_t)B_ * NSEL_ * C_;   // token_idx as floats (output #2)

  float* score_ws = (float*)d_ws;                                   // B_*TOK_ floats
  int*   tok_ws   = (int*)((char*)d_ws + (size_t)B_ * TOK_ * 4);    // B_*NSEL_ ints

  nepam_zero  <<<(B_ * TOK_ + 255) / 256, 256, 0, stream>>>(score_ws, B_ * TOK_);
  nepam_score <<<B_ * 8, 256, 0, stream>>>(x, score_ws);
  nepam_select<<<B_, 512, 0, stream>>>(score_ws, tok_ws, tok_out);
  nepam_gather<<<dim3(B_, NCC_), 256, CCH_ * LSTR_ * 4, stream>>>(x, tok_ws, out);
}